// GCN_VS_73555609911564
// MI455X (gfx1250) — compile-verified
//
#include <hip/hip_runtime.h>

typedef float v2f __attribute__((ext_vector_type(2)));
typedef float v8f __attribute__((ext_vector_type(8)));

#define D_K 128   // inner dim of every layer's GEMM (D_IN == D_H == 128)

// ---------------------------------------------------------------------------
// G[row, col] = dinv[row] * sum_k X[row,k] * W[k,col]
// One wave computes one 16x16 output tile via V_WMMA_F32_16X16X4_F32.
// Block = (DOUT/16) waves covering one 16-row strip; grid.x = N/16.
// DOUT is a template constant so every load in the K-loop uses an immediate
// IOFFSET (no per-iteration 64-bit address math).
// ---------------------------------------------------------------------------
template <int DOUT>
__global__ __launch_bounds__(256) void gcn_gemm_scale_kernel(
    const float* __restrict__ X, const float* __restrict__ W,
    const float* __restrict__ dinv, float* __restrict__ G)
{
    const int wave = threadIdx.x >> 5;
    const int lane = threadIdx.x & 31;
    const int rowBase = blockIdx.x << 4;
    const int hi  = lane >> 4;          // 0: lanes 0-15, 1: lanes 16-31
    const int l15 = lane & 15;
    const int col = (wave << 4) + l15;  // this wave's 16-col tile

    // Per-lane base pointers; hi*2 K-skew folded in once.
    const float* __restrict__ xbase = X + (size_t)(rowBase + l15) * D_K + (hi << 1);
    const float* __restrict__ wbase = W + (size_t)(hi << 1) * DOUT + col;

    // Warm GL2 with this wave's W panel (reused by all 6250 blocks).
    __builtin_prefetch(wbase, 0, 1);

    v8f acc = {0.f, 0.f, 0.f, 0.f, 0.f, 0.f, 0.f, 0.f};
#pragma unroll
    for (int k0 = 0; k0 < D_K; k0 += 4) {
        v2f a;                          // A 16x4: lanes 0-15 K=k0,k0+1 ; lanes 16-31 K=k0+2,k0+3
        a.x = xbase[k0];
        a.y = xbase[k0 + 1];
        v2f b;                          // B 4x16: VGPR0 rows K=k0 / k0+2, VGPR1 rows K=k0+1 / k0+3
        b.x = wbase[(size_t)k0 * DOUT];
        b.y = wbase[(size_t)(k0 + 1) * DOUT];
        acc = __builtin_amdgcn_wmma_f32_16x16x4_f32(
            /*neg_a=*/false, a, /*neg_b=*/false, b,
            /*c_mod=*/(short)0, acc, /*reuse_a=*/false, /*reuse_b=*/false);
    }

    // C/D layout: VGPR r -> M = r + 8*hi, N = l15
    const int row0 = rowBase + (hi << 3);
    float* __restrict__ gout = G + (size_t)row0 * DOUT + col;
    const float* __restrict__ dbase = dinv + row0;
#pragma unroll
    for (int r = 0; r < 8; ++r) {
        gout[(size_t)r * DOUT] = acc[r] * dbase[r];
    }
}

// ---------------------------------------------------------------------------
// Small elementwise helpers
// ---------------------------------------------------------------------------
__global__ void fill_kernel(float* __restrict__ p, float v, int n) {
    int i = blockIdx.x * blockDim.x + threadIdx.x;
    if (i < n) p[i] = v;
}

__global__ void deg_count_kernel(const int* __restrict__ dst,
                                 float* __restrict__ deg, int E) {
    int e = blockIdx.x * blockDim.x + threadIdx.x;
    if (e < E) unsafeAtomicAdd(&deg[dst[e]], 1.0f);
}

__global__ void rsqrt_inplace_kernel(float* __restrict__ p, int n) {
    int i = blockIdx.x * blockDim.x + threadIdx.x;
    if (i < n) p[i] = rsqrtf(p[i]);
}

__global__ void zero4_kernel(float4* __restrict__ p, int n4) {
    int i = blockIdx.x * blockDim.x + threadIdx.x;
    if (i < n4) p[i] = make_float4(0.f, 0.f, 0.f, 0.f);
}

// ---------------------------------------------------------------------------
// ACC[dst[e], :] += G[src[e], :]   (norm already folded into G by the GEMM)
// One thread moves one float4; SHIFT = log2(DOUT/4).
// ---------------------------------------------------------------------------
template <int SHIFT>
__global__ __launch_bounds__(256) void scatter_kernel(
    const float* __restrict__ G, const int* __restrict__ src,
    const int* __restrict__ dst, float* __restrict__ ACC, int E)
{
    const long long t = (long long)blockIdx.x * blockDim.x + threadIdx.x;
    const int e = (int)(t >> SHIFT);
    if (e >= E) return;
    const int f = (int)(t & ((1 << SHIFT) - 1)) << 2;   // feature offset (floats)
    constexpr int DOUT = 4 << SHIFT;
    const int s = src[e];
    const int d = dst[e];
    const float4 g = *reinterpret_cast<const float4*>(G + (size_t)s * DOUT + f);
    float* a = ACC + (size_t)d * DOUT + f;
    unsafeAtomicAdd(a + 0, g.x);
    unsafeAtomicAdd(a + 1, g.y);
    unsafeAtomicAdd(a + 2, g.z);
    unsafeAtomicAdd(a + 3, g.w);
}

// ---------------------------------------------------------------------------
// OUT[i,f] = [relu]( dinv[i] * (ACC[i,f] + G[i,f]) + b[f] )
// t enumerates all float4 slots: address offset = 4*t, node i = t >> SHIFT.
// ---------------------------------------------------------------------------
template <int SHIFT>
__global__ __launch_bounds__(256) void finalize_kernel(
    const float* __restrict__ G, const float* __restrict__ ACC,
    const float* __restrict__ dinv, const float* __restrict__ bias,
    float* __restrict__ OUT, int n4, int relu)
{
    const int t = blockIdx.x * blockDim.x + threadIdx.x;
    if (t >= n4) return;
    const int i = t >> SHIFT;
    const int f = (t & ((1 << SHIFT) - 1)) << 2;
    const float di = dinv[i];
    const float4 g = *reinterpret_cast<const float4*>(G   + (size_t)t * 4);
    const float4 a = *reinterpret_cast<const float4*>(ACC + (size_t)t * 4);
    const float4 b = *reinterpret_cast<const float4*>(bias + f);
    float4 o;
    o.x = di * (a.x + g.x) + b.x;
    o.y = di * (a.y + g.y) + b.y;
    o.z = di * (a.z + g.z) + b.z;
    o.w = di * (a.w + g.w) + b.w;
    if (relu) {
        o.x = fmaxf(o.x, 0.f);
        o.y = fmaxf(o.y, 0.f);
        o.z = fmaxf(o.z, 0.f);
        o.w = fmaxf(o.w, 0.f);
    }
    *reinterpret_cast<float4*>(OUT + (size_t)t * 4) = o;
}

// ---------------------------------------------------------------------------
// Host side
// inputs: x, edge_index, training, W1,b1, W2,b2, W3,b3, W4,b4
// ---------------------------------------------------------------------------
extern "C" void kernel_launch(void* const* d_in, const int* in_sizes, int n_in,
                              void* d_out, int out_size, void* d_ws, size_t ws_size,
                              hipStream_t stream)
{
    const float* x   = (const float*)d_in[0];
    const int*   ei  = (const int*)  d_in[1];   // [2, E] (src row, then dst row)
    const float* W1  = (const float*)d_in[3];
    const float* b1  = (const float*)d_in[4];
    const float* W2  = (const float*)d_in[5];
    const float* b2  = (const float*)d_in[6];
    const float* W3  = (const float*)d_in[7];
    const float* b3  = (const float*)d_in[8];
    const float* W4  = (const float*)d_in[9];
    const float* b4  = (const float*)d_in[10];

    const int N = in_sizes[0] / D_K;   // 100000 (divisible by 16)
    const int E = in_sizes[1] / 2;     // 600000
    const int* src = ei;
    const int* dst = ei + E;

    // Workspace layout (floats): dinv[N] | GA[N*128] | GB[N*128] | ACC[N*128]
    float* dinv = (float*)d_ws;
    float* GA   = dinv + N;
    float* GB   = GA + (size_t)N * D_K;
    float* ACC  = GB + (size_t)N * D_K;

    // Degrees (once): dinv = rsqrt(1 + indeg)
    fill_kernel<<<(N + 255) / 256, 256, 0, stream>>>(dinv, 1.0f, N);
    deg_count_kernel<<<(E + 255) / 256, 256, 0, stream>>>(dst, dinv, E);
    rsqrt_inplace_kernel<<<(N + 255) / 256, 256, 0, stream>>>(dinv, N);

    // ---- layers with DOUT = 128 ----
    auto run128 = [&](const float* Xin, const float* W, const float* b,
                      float* G, float* Out) {
        gcn_gemm_scale_kernel<128><<<N / 16, 256, 0, stream>>>(Xin, W, dinv, G);
        const int n4 = N * 32;
        zero4_kernel<<<(n4 + 255) / 256, 256, 0, stream>>>((float4*)ACC, n4);
        const long long e4 = (long long)E * 32;
        scatter_kernel<5><<<(int)((e4 + 255) / 256), 256, 0, stream>>>(G, src, dst, ACC, E);
        finalize_kernel<5><<<(n4 + 255) / 256, 256, 0, stream>>>(G, ACC, dinv, b, Out, n4, 1);
    };

    // finalize writes in-place over G; that buffer becomes the next layer's X.
    run128(x,  W1, b1, GA, GA);
    run128(GA, W2, b2, GB, GB);
    run128(GB, W3, b3, GA, GA);

    // ---- last layer, DOUT = 64, no ReLU, straight to d_out ----
    {
        gcn_gemm_scale_kernel<64><<<N / 16, 128, 0, stream>>>(GA, W4, dinv, GB);
        const int n4 = N * 16;
        zero4_kernel<<<(n4 + 255) / 256, 256, 0, stream>>>((float4*)ACC, n4);
        const long long e4 = (long long)E * 16;
        scatter_kernel<4><<<(int)((e4 + 255) / 256), 256, 0, stream>>>(GB, src, dst, ACC, E);
        finalize_kernel<4><<<(n4 + 255) / 256, 256, 0, stream>>>(GB, ACC, dinv, b4,
                                                                 (float*)d_out, n4, 0);
    }
}